// GTN_Rec_15358803050622
// MI455X (gfx1250) — compile-verified
//
#include <hip/hip_runtime.h>
#include <math.h>

typedef __attribute__((ext_vector_type(2))) float v2f;
typedef __attribute__((ext_vector_type(8))) float v8f;
typedef __attribute__((ext_vector_type(4))) unsigned int v4u;
typedef __attribute__((ext_vector_type(8))) int v8i;
typedef __attribute__((ext_vector_type(4))) int v4i;

#define NN      2000   // nodes
#define EE      3      // edge types
#define BB      64     // batch
#define SS      30     // seq len
#define DD      128    // embed dim
#define UU      128    // hidden dim
#define MROWS   (BB*SS)  // 1920

// GEMM tiling
#define BM 128
#define BN 128
#define BK 32
#define GEMM_THREADS 256

#if defined(__HIP_DEVICE_COMPILE__) && __has_builtin(__builtin_amdgcn_tensor_load_to_lds) && __has_builtin(__builtin_amdgcn_s_wait_tensorcnt)
#define USE_TDM 1
#else
#define USE_TDM 0
#endif

// ---------------------------------------------------------------------------
// TDM copy of a dense fp32 tile (128 rows x 32 cols) of a row-major matrix
// into LDS laid out as [row][BK+1] (pad 1 dword per 32 -> bank-conflict-free).
// OOB rows/cols (beyond validRows/validCols) read as zero per TDM semantics.
// Must be called from a wave-uniform context (one wave only).
// ---------------------------------------------------------------------------
#if USE_TDM
__device__ __forceinline__ void tdm_load_tile_a(float* ldsDst,
                                                const float* gsrc,
                                                int validCols, int validRows,
                                                int ldaRow) {
  const unsigned long long ga = (unsigned long long)(uintptr_t)gsrc;
  v4u g0;
  g0.x = 1u;                                        // count=1, no gather
  g0.y = (unsigned int)(uintptr_t)ldsDst;           // LDS byte offset
  g0.z = (unsigned int)(ga & 0xFFFFFFFFull);        // global_addr[31:0]
  g0.w = ((unsigned int)(ga >> 32) & 0x01FFFFFFu)   // global_addr[56:32]
         | (2u << 30);                              // type = 2 ("image")
  v8i g1;
  // wg_mask=0 | data_size=4B(2) | pad_enable | pad_interval=32dw(4) | pad=1dw(0)
  g1[0] = (2 << 16) | (1 << 20) | (4 << 22);
  g1[1] = (int)(((unsigned)validCols & 0xFFFFu) << 16);          // dim0 lo16
  g1[2] = (int)((((unsigned)validCols >> 16) & 0xFFFFu)
                | (((unsigned)validRows & 0xFFFFu) << 16));      // dim0 hi|dim1 lo
  g1[3] = (int)((((unsigned)validRows >> 16) & 0xFFFFu)
                | (32u << 16));                                  // dim1 hi|tile_dim0
  g1[4] = (int)(128u);                                           // tile_dim1 (tile_dim2=0)
  g1[5] = ldaRow;                                                // dim0_stride lo32
  g1[6] = 0;                                                     // stride hi / dim1_stride lo
  g1[7] = 0;
  const v4i gz4 = {0, 0, 0, 0};                    // 2D: groups 2/3 unused
  const v8i gz8 = {0, 0, 0, 0, 0, 0, 0, 0};
  __builtin_amdgcn_tensor_load_to_lds(g0, g1, gz4, gz4, gz8, 0);
}
#endif

// ---------------------------------------------------------------------------
// Softmax of channel-0 rows of W1a / W1b / W2 -> 9 mix weights in ws
// ---------------------------------------------------------------------------
__global__ void mix_weights_kernel(const float* __restrict__ W1a,
                                   const float* __restrict__ W1b,
                                   const float* __restrict__ W2,
                                   float* __restrict__ w9) {
  if (threadIdx.x == 0) {
    const float* Ws[3] = {W1a, W1b, W2};
    for (int m = 0; m < 3; ++m) {
      const float* r = Ws[m];                 // row 0 == channel 0, E=3
      float mx = fmaxf(r[0], fmaxf(r[1], r[2]));
      float e0 = __expf(r[0] - mx);
      float e1 = __expf(r[1] - mx);
      float e2 = __expf(r[2] - mx);
      float inv = 1.0f / (e0 + e1 + e2);
      w9[m * 3 + 0] = e0 * inv;
      w9[m * 3 + 1] = e1 * inv;
      w9[m * 3 + 2] = e2 * inv;
    }
  }
}

// ---------------------------------------------------------------------------
// C(M x N) = Asrc(M x K) @ B where B[k][n] = sum_e w3[e] * Adj[(k*N+n)*3+e]
// EPI==0: plain store.
// EPI==1: C = X*relu(iB) + relu(P - thr)   (basket encoding fuse)
// fp32 WMMA 16x16x4; block tile 128x128x32; 8 waves, each wave 64x32.
// A tile staged via TDM (tensor_load_to_lds) with 1-dword/row LDS padding.
// ---------------------------------------------------------------------------
template <int EPI>
__global__ __launch_bounds__(GEMM_THREADS)
void gemm_mixb_kernel(const float* __restrict__ Asrc, int M, int lda,
                      const float* __restrict__ Adj, int N, int K,
                      const float* __restrict__ w3,
                      float* __restrict__ C, int ldc,
                      const float* __restrict__ Xepi,
                      const float* __restrict__ iB,
                      const float* __restrict__ thr) {
  __shared__ float As[BM][BK + 1];   // [m][k], +1 pad dword per row
  __shared__ float Bs[BK][BN + 4];   // [k][n]

  const int tid  = threadIdx.x;
  const int lane = tid & 31;
  const int wave = tid >> 5;        // 0..7
  const int waveM = wave >> 2;      // 0..1  (64-row slab)
  const int waveN = wave & 3;       // 0..3  (32-col slab)
  const int l16 = lane & 15;
  const int hi  = lane >> 4;        // 0/1: upper half-lanes

  const int row0 = blockIdx.y * BM;
  const int col0 = blockIdx.x * BN;

  const float w0 = w3[0], w1 = w3[1], w2 = w3[2];

  v8f acc[4][2];
  const v8f vzero = {0.f, 0.f, 0.f, 0.f, 0.f, 0.f, 0.f, 0.f};
#pragma unroll
  for (int mt = 0; mt < 4; ++mt)
#pragma unroll
    for (int nt = 0; nt < 2; ++nt) acc[mt][nt] = vzero;

  for (int kk = 0; kk < K; kk += BK) {
    // ---- stage A tile into As[m][k] ----
#if USE_TDM
    if (wave == 0) {
      tdm_load_tile_a(&As[0][0], Asrc + (size_t)row0 * lda + kk,
                      K - kk, M - row0, lda);
    }
#else
    {
      const int rowIdx = tid >> 3;          // 0..31
      const int kIdx   = (tid & 7) * 4;     // 0,4,..,28
#pragma unroll
      for (int p = 0; p < 4; ++p) {
        const int m  = rowIdx + p * 32;
        const int gr = row0 + m;
#pragma unroll
        for (int j = 0; j < 4; ++j) {
          const int gk = kk + kIdx + j;
          float v = 0.f;
          if (gr < M && gk < K) v = Asrc[(size_t)gr * lda + gk];
          As[m][kIdx + j] = v;
        }
      }
    }
#endif
    // ---- stage B tile: mix 3 adjacency slices on the fly ----
    {
      const int n     = tid & 127;          // 0..127 (coalesced in n)
      const int kBase = tid >> 7;           // 0..1
#pragma unroll
      for (int p = 0; p < 16; ++p) {
        const int k  = kBase + p * 2;
        const int gk = kk + k;
        const int gn = col0 + n;
        float v = 0.f;
        if (gk < K && gn < N) {
          const float* ap = Adj + ((size_t)gk * N + gn) * EE;
          v = w0 * ap[0] + w1 * ap[1] + w2 * ap[2];
        }
        Bs[k][n] = v;
      }
      // prefetch next chunk of adjacency into caches (global_prefetch_b8)
      if (kk + BK < K) {
        const float* pf = Adj + ((size_t)(kk + BK + kBase) * N + (col0 + n)) * EE;
        __builtin_prefetch(pf, 0, 1);
      }
    }
#if USE_TDM
    if (wave == 0) __builtin_amdgcn_s_wait_tensorcnt(0);
#endif
    __syncthreads();

    // ---- 8 K-steps of fp32 WMMA (K=4 each) ----
#pragma unroll
    for (int k4 = 0; k4 < BK; k4 += 4) {
      const int ka = k4 + hi * 2;           // lanes 16-31 carry K+2 (ISA layout)
      v2f afr[4], bfr[2];
#pragma unroll
      for (int mt = 0; mt < 4; ++mt) {
        const int m = waveM * 64 + mt * 16 + l16;
        afr[mt].x = As[m][ka + 0];
        afr[mt].y = As[m][ka + 1];
      }
#pragma unroll
      for (int nt = 0; nt < 2; ++nt) {
        const int n = waveN * 32 + nt * 16 + l16;
        bfr[nt].x = Bs[ka + 0][n];
        bfr[nt].y = Bs[ka + 1][n];
      }
#pragma unroll
      for (int mt = 0; mt < 4; ++mt)
#pragma unroll
        for (int nt = 0; nt < 2; ++nt)
          acc[mt][nt] = __builtin_amdgcn_wmma_f32_16x16x4_f32(
              false, afr[mt], false, bfr[nt], (short)0, acc[mt][nt],
              false, false);
    }
    __syncthreads();
  }

  // ---- epilogue + store (C/D layout: VGPR r -> M=r (lanes 0-15) / r+8) ----
#pragma unroll
  for (int mt = 0; mt < 4; ++mt) {
#pragma unroll
    for (int nt = 0; nt < 2; ++nt) {
      const int n = col0 + waveN * 32 + nt * 16 + l16;
      if (n >= N) continue;
#pragma unroll
      for (int vr = 0; vr < 8; ++vr) {
        const int m = row0 + waveM * 64 + mt * 16 + hi * 8 + vr;
        if (m >= M) continue;
        float v = acc[mt][nt][vr];
        if (EPI == 1) {
          const float sc = fmaxf(iB[n], 0.f);
          const float x  = Xepi[(size_t)m * N + n];
          v = x * sc + fmaxf(v - thr[0], 0.f);
        }
        C[(size_t)m * ldc + n] = v;
      }
    }
  }
}

// ---------------------------------------------------------------------------
// basket = relu( enc(M x K) @ Wt(N x K)^T + bias )   (N = 128)
// ---------------------------------------------------------------------------
__global__ __launch_bounds__(GEMM_THREADS)
void gemm_btrans_relu_kernel(const float* __restrict__ Asrc, int M, int K,
                             const float* __restrict__ Wt,   // (N x K) row-major
                             const float* __restrict__ bias, // (N)
                             int N,
                             float* __restrict__ C) {        // (M x N)
  __shared__ float As[BM][BK + 1];
  __shared__ float Bs[BK][BN + 4];

  const int tid  = threadIdx.x;
  const int lane = tid & 31;
  const int wave = tid >> 5;
  const int waveM = wave >> 2;
  const int waveN = wave & 3;
  const int l16 = lane & 15;
  const int hi  = lane >> 4;

  const int row0 = blockIdx.y * BM;
  const int col0 = blockIdx.x * BN;   // 0 (N==128)

  v8f acc[4][2];
  const v8f vzero = {0.f, 0.f, 0.f, 0.f, 0.f, 0.f, 0.f, 0.f};
#pragma unroll
  for (int mt = 0; mt < 4; ++mt)
#pragma unroll
    for (int nt = 0; nt < 2; ++nt) acc[mt][nt] = vzero;

  for (int kk = 0; kk < K; kk += BK) {
#if USE_TDM
    if (wave == 0) {
      tdm_load_tile_a(&As[0][0], Asrc + (size_t)row0 * K + kk,
                      K - kk, M - row0, K);
    }
#else
    {
      const int rowIdx = tid >> 3;
      const int kIdx   = (tid & 7) * 4;
#pragma unroll
      for (int p = 0; p < 4; ++p) {
        const int m  = rowIdx + p * 32;
        const int gr = row0 + m;
#pragma unroll
        for (int j = 0; j < 4; ++j) {
          const int gk = kk + kIdx + j;
          float v = 0.f;
          if (gr < M && gk < K) v = Asrc[(size_t)gr * K + gk];
          As[m][kIdx + j] = v;
        }
      }
    }
#endif
    {
      const int n    = tid >> 1;          // 0..127
      const int kIdx = (tid & 1) * 16;    // 0/16
#pragma unroll
      for (int j = 0; j < 16; ++j) {
        const int gk = kk + kIdx + j;
        const int gn = col0 + n;
        float v = 0.f;
        if (gk < K && gn < N) v = Wt[(size_t)gn * K + gk];
        Bs[kIdx + j][n] = v;
      }
    }
#if USE_TDM
    if (wave == 0) __builtin_amdgcn_s_wait_tensorcnt(0);
#endif
    __syncthreads();

#pragma unroll
    for (int k4 = 0; k4 < BK; k4 += 4) {
      const int ka = k4 + hi * 2;
      v2f afr[4], bfr[2];
#pragma unroll
      for (int mt = 0; mt < 4; ++mt) {
        const int m = waveM * 64 + mt * 16 + l16;
        afr[mt].x = As[m][ka + 0];
        afr[mt].y = As[m][ka + 1];
      }
#pragma unroll
      for (int nt = 0; nt < 2; ++nt) {
        const int n = waveN * 32 + nt * 16 + l16;
        bfr[nt].x = Bs[ka + 0][n];
        bfr[nt].y = Bs[ka + 1][n];
      }
#pragma unroll
      for (int mt = 0; mt < 4; ++mt)
#pragma unroll
        for (int nt = 0; nt < 2; ++nt)
          acc[mt][nt] = __builtin_amdgcn_wmma_f32_16x16x4_f32(
              false, afr[mt], false, bfr[nt], (short)0, acc[mt][nt],
              false, false);
    }
    __syncthreads();
  }

#pragma unroll
  for (int mt = 0; mt < 4; ++mt) {
#pragma unroll
    for (int nt = 0; nt < 2; ++nt) {
      const int n = col0 + waveN * 32 + nt * 16 + l16;
      if (n >= N) continue;
      const float bn = bias[n];
#pragma unroll
      for (int vr = 0; vr < 8; ++vr) {
        const int m = row0 + waveM * 64 + mt * 16 + hi * 8 + vr;
        if (m >= M) continue;
        C[(size_t)m * N + n] = fmaxf(acc[mt][nt][vr] + bn, 0.f);
      }
    }
  }
}

// ---------------------------------------------------------------------------
// LSTM: single workgroup (512 threads = 16 waves). h in LDS, c in registers.
// gates order i,f,g,o (torch). Captures last = h[t] at t == seq_len[b]-1.
// ---------------------------------------------------------------------------
__device__ __forceinline__ float dsigmoid(float x) {
  return 1.0f / (1.0f + __expf(-x));
}

__global__ __launch_bounds__(512)
void lstm_kernel(const float* __restrict__ basket,  // (B,S,D)
                 const float* __restrict__ Wih,     // (4U, D)
                 const float* __restrict__ Whh,     // (4U, U)
                 const float* __restrict__ bih,
                 const float* __restrict__ bhh,
                 const float* __restrict__ h0,      // (1,B,U)
                 const float* __restrict__ c0,
                 const int* __restrict__ seq_len,   // (B)
                 float* __restrict__ lastb) {       // (B,U)
  __shared__ float hS[BB * UU];
  __shared__ float biasS[4 * UU];

  const int tid = threadIdx.x;
  for (int i = tid; i < BB * UU; i += 512) hS[i] = h0[i];
  for (int i = tid; i < 4 * UU; i += 512) biasS[i] = bih[i] + bhh[i];

  float creg[16], nh[16];
  const int base = tid * 16;                // 512*16 = 8192 = B*U
#pragma unroll
  for (int j = 0; j < 16; ++j) creg[j] = c0[base + j];
  __syncthreads();

  for (int t = 0; t < SS; ++t) {
    for (int j = 0; j < 16; ++j) {
      const int idx = base + j;
      const int b = idx >> 7;
      const int u = idx & 127;
      const float* xrow = basket + ((size_t)(b * SS + t)) * DD;
      const float* hrow = hS + b * UU;
      float gi = biasS[u], gf = biasS[UU + u], gg = biasS[2 * UU + u],
            go = biasS[3 * UU + u];
      const float* wi = Wih + (size_t)u * DD;
      const float* wf = Wih + (size_t)(UU + u) * DD;
      const float* wg = Wih + (size_t)(2 * UU + u) * DD;
      const float* wo = Wih + (size_t)(3 * UU + u) * DD;
      for (int d = 0; d < DD; ++d) {
        const float x = xrow[d];
        gi += x * wi[d]; gf += x * wf[d]; gg += x * wg[d]; go += x * wo[d];
      }
      const float* vi = Whh + (size_t)u * UU;
      const float* vf = Whh + (size_t)(UU + u) * UU;
      const float* vg = Whh + (size_t)(2 * UU + u) * UU;
      const float* vo = Whh + (size_t)(3 * UU + u) * UU;
      for (int d = 0; d < UU; ++d) {
        const float h = hrow[d];
        gi += h * vi[d]; gf += h * vf[d]; gg += h * vg[d]; go += h * vo[d];
      }
      const float ig = dsigmoid(gi), fg = dsigmoid(gf);
      const float gg2 = tanhf(gg),   og = dsigmoid(go);
      const float cc = fg * creg[j] + ig * gg2;
      creg[j] = cc;
      nh[j] = og * tanhf(cc);
      if (t == seq_len[b] - 1) lastb[idx] = nh[j];
    }
    __syncthreads();                       // all reads of old h done
#pragma unroll
    for (int j = 0; j < 16; ++j) hS[base + j] = nh[j];
    __syncthreads();
  }
}

// ---------------------------------------------------------------------------
// predict = (1-a)*sigmoid(last @ Wscore^T) + a*(... * relu(I_B))
// ---------------------------------------------------------------------------
__global__ void score_kernel(const float* __restrict__ lastb,   // (B,U)
                             const float* __restrict__ Wscore,  // (N,U)
                             const float* __restrict__ iB,      // (N)
                             float* __restrict__ out) {         // (B,N)
  const int idx = blockIdx.x * blockDim.x + threadIdx.x;
  if (idx >= BB * NN) return;
  const int b = idx / NN;
  const int n = idx - b * NN;
  const float* l = lastb + b * UU;
  const float* w = Wscore + (size_t)n * UU;
  float s = 0.f;
#pragma unroll 8
  for (int u = 0; u < UU; ++u) s += l[u] * w[u];
  const float p  = dsigmoid(s);
  const float sc = fmaxf(iB[n], 0.f);
  out[idx] = 0.5f * p + 0.5f * (p * sc);   // ALPHA = 0.5
}

// ---------------------------------------------------------------------------
extern "C" void kernel_launch(void* const* d_in, const int* in_sizes, int n_in,
                              void* d_out, int out_size, void* d_ws,
                              size_t ws_size, hipStream_t stream) {
  const float* A      = (const float*)d_in[0];   // (N,N,E)
  const int*   seqlen = (const int*)d_in[1];     // (B)
  const float* seqs   = (const float*)d_in[2];   // (B,S,N)
  const float* h0     = (const float*)d_in[3];
  const float* c0     = (const float*)d_in[4];
  const float* W1a    = (const float*)d_in[5];
  const float* W1b    = (const float*)d_in[6];
  const float* W2     = (const float*)d_in[7];
  const float* lin_w  = (const float*)d_in[8];   // (D,N)
  const float* lin_b  = (const float*)d_in[9];   // (D)
  const float* Wih    = (const float*)d_in[10];
  const float* Whh    = (const float*)d_in[11];
  const float* bih    = (const float*)d_in[12];
  const float* bhh    = (const float*)d_in[13];
  const float* Wsc    = (const float*)d_in[14];  // (N,U)
  const float* iB     = (const float*)d_in[15];  // (N)
  const float* thr    = (const float*)d_in[16];  // (1)

  // workspace layout (everything read is written first each call)
  const size_t SLOT = (size_t)MROWS * NN * sizeof(float);  // 15,360,000 B
  char*  ws    = (char*)d_ws;
  float* w9    = (float*)ws;                  // 9 mix weights
  float* slotA = (float*)(ws + 256);          // P1 -> enc
  float* slotB = (float*)(ws + 256 + SLOT);   // P2 -> basket
  float* lastb = (float*)(ws + 256 + 2 * SLOT);  // (B,U)

  mix_weights_kernel<<<1, 32, 0, stream>>>(W1a, W1b, W2, w9);

  const dim3 blk(GEMM_THREADS);
  const dim3 grid((NN + BN - 1) / BN, (MROWS + BM - 1) / BM);  // (16, 15)

  // P1 = x @ mix(A, wa)
  gemm_mixb_kernel<0><<<grid, blk, 0, stream>>>(
      seqs, MROWS, NN, A, NN, NN, w9 + 0, slotA, NN, nullptr, nullptr, nullptr);
  // P2 = P1 @ mix(A, wb)
  gemm_mixb_kernel<0><<<grid, blk, 0, stream>>>(
      slotA, MROWS, NN, A, NN, NN, w9 + 3, slotB, NN, nullptr, nullptr, nullptr);
  // enc = x*relu(I_B) + relu(P2 @ mix(A, w2) - thr)
  gemm_mixb_kernel<1><<<grid, blk, 0, stream>>>(
      slotB, MROWS, NN, A, NN, NN, w9 + 6, slotA, NN, seqs, iB, thr);
  // basket = relu(enc @ lin_w^T + lin_b)
  gemm_btrans_relu_kernel<<<dim3(1, (MROWS + BM - 1) / BM), blk, 0, stream>>>(
      slotA, MROWS, NN, lin_w, lin_b, DD, slotB);
  // LSTM -> last hidden per batch
  lstm_kernel<<<1, 512, 0, stream>>>(slotB, Wih, Whh, bih, bhh, h0, c0,
                                     seqlen, lastb);
  // scores -> d_out (B,N) fp32
  score_kernel<<<(BB * NN + 255) / 256, 256, 0, stream>>>(lastb, Wsc, iB,
                                                          (float*)d_out);
}